// CoAttentionLayer_14465449853161
// MI455X (gfx1250) — compile-verified
//
#include <hip/hip_runtime.h>
#include <hip/hip_bf16.h>

typedef __attribute__((ext_vector_type(2))) float v2f;
typedef __attribute__((ext_vector_type(8))) float v8f;

#define WMMA_F32(a, b, c) \
    __builtin_amdgcn_wmma_f32_16x16x4_f32(false, (a), false, (b), (short)0, (c), false, false)

constexpr int Bn  = 32;
constexpr int Cc  = 256;
constexpr int HW  = 1024;   // 32*32
constexpr int HID = 64;

constexpr int SROW = 1028;  // 1024 + 4 pad: kills bank conflicts on column reads
constexpr int QROW = 68;    // 64 + 4 pad

// ---------------------------------------------------------------------------
// Kernel 1: 1x1-conv projections via V_WMMA_F32_16X16X4_F32.
// mode 0: Qp[b][pix][hid] = Xq^T * Wq^T + bq   (D[m=pix][n=hid])
// mode 1: Kp[b][hid][pix] = Wk   * Xk   + bk   (D[m=hid][n=pix])
// ---------------------------------------------------------------------------
__global__ __launch_bounds__(256) void coattn_proj_kernel(
    const float* __restrict__ q_feat, const float* __restrict__ k_feat,
    const float* __restrict__ Wq, const float* __restrict__ bq,
    const float* __restrict__ Wk, const float* __restrict__ bk,
    const int*   __restrict__ same_WqWk,
    float* __restrict__ Qp, float* __restrict__ Kp)
{
    const int lane = threadIdx.x & 31;
    const int wave = threadIdx.x >> 5;
    const int half = lane >> 4;        // which 16-lane half of the wave
    const int l16  = lane & 15;
    const int b    = blockIdx.y;
    const int mode = blockIdx.z;
    const int p0   = blockIdx.x * 128 + wave * 16;   // 16-pixel tile per wave

    const float* X;  const float* W;  const float* bias;
    if (mode == 0) {
        X = q_feat + (size_t)b * Cc * HW;  W = Wq;  bias = bq;
    } else {
        X = k_feat + (size_t)b * Cc * HW;
        const bool same = (*same_WqWk) != 0;
        W = same ? Wq : Wk;  bias = same ? bq : bk;
    }

    for (int t = 0; t < 4; ++t) {             // hid tile: 4 x 16 = 64
        const int h0 = t * 16;
        v8f acc = {};
        if (mode == 0) {
            // A[m=pix][k=c] from X (coalesced b32 x2), B[k=c][n=hid] = W^T (b64/lane)
            for (int c0 = 0; c0 < Cc; c0 += 4) {
                v2f a, bf;
                a.x = X[(size_t)(c0 + 2 * half    ) * HW + p0 + l16];
                a.y = X[(size_t)(c0 + 2 * half + 1) * HW + p0 + l16];
                const float* wrow = W + (size_t)(h0 + l16) * Cc + c0 + 2 * half;
                bf.x = wrow[0];  bf.y = wrow[1];
                acc = WMMA_F32(a, bf, acc);
            }
            const float bv = bias[h0 + l16];          // n = hid column
            float* dst = Qp + ((size_t)b * HW + p0) * HID + h0 + l16;
#pragma unroll
            for (int v = 0; v < 8; ++v)
                dst[(size_t)(v + 8 * half) * HID] = acc[v] + bv;
        } else {
            // A[m=hid][k=c] = W (b64/lane), B[k=c][n=pix] from X (coalesced)
            for (int c0 = 0; c0 < Cc; c0 += 4) {
                v2f a, bf;
                const float* wrow = W + (size_t)(h0 + l16) * Cc + c0 + 2 * half;
                a.x = wrow[0];  a.y = wrow[1];
                bf.x = X[(size_t)(c0 + 2 * half    ) * HW + p0 + l16];
                bf.y = X[(size_t)(c0 + 2 * half + 1) * HW + p0 + l16];
                acc = WMMA_F32(a, bf, acc);
            }
            float* dst = Kp + ((size_t)b * HID + h0) * HW + p0 + l16;
#pragma unroll
            for (int v = 0; v < 8; ++v)               // m = hid row -> per-v bias
                dst[(size_t)(v + 8 * half) * HW] = acc[v] + bias[h0 + v + 8 * half];
        }
    }
}

// ---------------------------------------------------------------------------
// Kernel 2: fused S = Q^T K, softmax, O = V P^T for one (batch, 16-query tile).
// 8 waves. LDS holds the 16 x 1024 score slab (padded) + Q tile + 1/rowsum.
// ---------------------------------------------------------------------------
__global__ __launch_bounds__(256) void coattn_attend_kernel(
    const float* __restrict__ k_feat,   // V = key_features [B][C][HW]
    const float* __restrict__ Qp,       // [B][HW][HID]
    const float* __restrict__ Kp,       // [B][HID][HW]
    float* __restrict__ out)            // [B][C][HW]
{
    extern __shared__ float smem[];
    float* S   = smem;                    // 16 * SROW
    float* Qs  = smem + 16 * SROW;        // 16 * QROW
    float* inv = Qs + 16 * QROW;          // 16

    const int tid  = threadIdx.x;
    const int lane = tid & 31;
    const int wave = tid >> 5;
    const int half = lane >> 4;
    const int l16  = lane & 15;
    const int b    = blockIdx.y;
    const int q0   = blockIdx.x * 16;     // 16 query pixels per workgroup

    // ---- phase 0: stage Q tile [16 x 64] into LDS (4 floats / thread) ----
    {
        const int row  = tid >> 4;
        const int col4 = (tid & 15) * 4;
        const float* src = Qp + ((size_t)b * HW + q0 + row) * HID + col4;
        float* dst = Qs + row * QROW + col4;
#pragma unroll
        for (int i = 0; i < 4; ++i) dst[i] = src[i];
    }
    __syncthreads();

    // ---- phase 1: S[16 x 1024] = Q_tile(16x64) * K(64x1024), WMMA over K=64 --
    for (int kt = wave; kt < HW / 16; kt += 8) {
        v8f acc = {};
        const float* kp = Kp + (size_t)b * HID * HW + kt * 16 + l16;
        const float* qrowp = Qs + l16 * QROW + 2 * half;
        for (int h0 = 0; h0 < HID; h0 += 4) {
            v2f a, bf;
            a.x  = qrowp[h0];                           // A[m=q][k=h] (ds b64)
            a.y  = qrowp[h0 + 1];
            bf.x = kp[(size_t)(h0 + 2 * half    ) * HW]; // B[k=h][n=kpix]
            bf.y = kp[(size_t)(h0 + 2 * half + 1) * HW];
            acc = WMMA_F32(a, bf, acc);
        }
        float* srow = S + (8 * half) * SROW + kt * 16 + l16;
#pragma unroll
        for (int v = 0; v < 8; ++v) srow[v * SROW] = acc[v];
    }
    __syncthreads();

    // ---- phase 2: softmax over each of 16 rows (2 rows / wave) ----
    for (int r = wave * 2; r < wave * 2 + 2; ++r) {
        float* row = S + r * SROW;
        float m = -3.402823466e38f;
        for (int j = lane; j < HW; j += 32) m = fmaxf(m, row[j]);
#pragma unroll
        for (int off = 16; off > 0; off >>= 1) m = fmaxf(m, __shfl_xor(m, off, 32));
        float s = 0.f;
        for (int j = lane; j < HW; j += 32) {
            const float e = __expf(row[j] - m);
            row[j] = e;
            s += e;
        }
#pragma unroll
        for (int off = 16; off > 0; off >>= 1) s += __shfl_xor(s, off, 32);
        if (lane == 0) inv[r] = 1.0f / s;
    }
    __syncthreads();

    // ---- phase 3: O[c, q] = sum_p V[c,p] P[q,p], scale by 1/rowsum(q) ----
    const float* V = k_feat + (size_t)b * Cc * HW;
    for (int mt = wave * 2; mt < wave * 2 + 2; ++mt) {   // 2 channel tiles / wave
        const int c0 = mt * 16;
        v8f acc = {};
        const float* vrow = V + (size_t)(c0 + l16) * HW + 2 * half;  // A[m=c][k=p]
        const float* prow = S + l16 * SROW + 2 * half;               // B[k=p][n=q]
        for (int p0 = 0; p0 < HW; p0 += 4) {
            v2f a, bf;
            a.x  = vrow[p0];      a.y  = vrow[p0 + 1];   // global b64 / lane
            bf.x = prow[p0];      bf.y = prow[p0 + 1];   // ds b64 / lane
            acc = WMMA_F32(a, bf, acc);
        }
        const float scale = inv[l16];                    // n = query column
        float* dst = out + ((size_t)b * Cc + c0) * HW + q0 + l16;
#pragma unroll
        for (int v = 0; v < 8; ++v)
            dst[(size_t)(v + 8 * half) * HW] = acc[v] * scale;
    }
}

// ---------------------------------------------------------------------------
extern "C" void kernel_launch(void* const* d_in, const int* in_sizes, int n_in,
                              void* d_out, int out_size, void* d_ws, size_t ws_size,
                              hipStream_t stream) {
    const float* q_feat = (const float*)d_in[0];
    const float* k_feat = (const float*)d_in[1];
    const float* Wq     = (const float*)d_in[2];
    const float* bq     = (const float*)d_in[3];
    const float* Wk     = (const float*)d_in[4];
    const float* bk     = (const float*)d_in[5];
    // d_in[6] = vis_CA (unused by the math), d_in[7] = same_WqWk
    const int* same_WqWk = (const int*)d_in[7];

    float* Qp = (float*)d_ws;                         // 32*1024*64 f32 = 8 MB
    float* Kp = Qp + (size_t)Bn * HW * HID;           // next 8 MB

    coattn_proj_kernel<<<dim3(HW / 128, Bn, 2), 256, 0, stream>>>(
        q_feat, k_feat, Wq, bq, Wk, bk, same_WqWk, Qp, Kp);

    const size_t smem_bytes = (size_t)(16 * SROW + 16 * QROW + 16) * sizeof(float);
    coattn_attend_kernel<<<dim3(HW / 16, Bn), 256, smem_bytes, stream>>>(
        k_feat, Qp, Kp, (float*)d_out);
}